// ProposedModel_7971459301904
// MI455X (gfx1250) — compile-verified
//
#include <hip/hip_runtime.h>
#include <hip/hip_bf16.h>
#include <math.h>

// ---------------------------------------------------------------------------
// Problem constants (from reference)
// ---------------------------------------------------------------------------
#define D_        256            // emb dim
#define T_        24             // visits
#define NT_       23             // T-1 output rows
#define MP_       32             // NT_ padded to a multiple of 16 (WMMA M)
#define V2_       500            // med vocab / output classes
#define V2P_      512            // V2 padded to a multiple of 16 (WMMA N)
#define ILEN_     4000           // padded visit length; also pad index
#define G3_       768            // 3*D (GRU gates)
#define LM_       25             // meds per visit

typedef float v2f __attribute__((ext_vector_type(2)));
typedef float v8f __attribute__((ext_vector_type(8)));

// ---------------------------------------------------------------------------
// K0a: zero the M-padding rows (23..31) of the three WMMA A-operand buffers so
// the GEMM hot loop needs no row guards (no EXEC-mask branches around loads).
// grid = 27 blocks: region = b/9 (0:v_emb 1:h2 2:x_buf), row = 23 + b%9.
// ---------------------------------------------------------------------------
__global__ __launch_bounds__(D_)
void pad_zero_kernel(float* __restrict__ v_emb, float* __restrict__ h2,
                     float* __restrict__ x_buf)
{
    const int region = blockIdx.x / 9;
    const int row    = NT_ + (blockIdx.x - region * 9);
    float* p = (region == 0) ? v_emb : (region == 1) ? h2 : x_buf;
    p[row * D_ + threadIdx.x] = 0.0f;
}

// ---------------------------------------------------------------------------
// K0b: stage w_out into a zero-padded [512,256] copy so the pred GEMM needs no
// column guard in the hot loop. 0.5 MB, stays L2-hot.
// ---------------------------------------------------------------------------
__global__ __launch_bounds__(D_)
void pad_wout_kernel(const float* __restrict__ w_out, float* __restrict__ wpad)
{
    const int r = blockIdx.x, c = threadIdx.x;
    wpad[r * D_ + c] = (r < V2_) ? w_out[r * D_ + c] : 0.0f;
}

// ---------------------------------------------------------------------------
// K1: labels. Each of the 23*500 outputs scans its med row (25 entries): no
// scatter, no zero-init pass, trivially deterministic.
// ---------------------------------------------------------------------------
__global__ __launch_bounds__(256)
void build_labels_kernel(const int* __restrict__ med,
                         float* __restrict__ y1, int* __restrict__ y2)
{
    int i = blockIdx.x * blockDim.x + threadIdx.x;
    if (i >= NT_ * V2_) return;
    int t = i / V2_, n = i - t * V2_;
    const int* m = med + (t + 1) * LM_;
    float v = 0.0f;
#pragma unroll
    for (int j = 0; j < LM_; ++j) v = (m[j] == n) ? 1.0f : v;
    y1[i] = v;
    y2[i] = (n < LM_) ? m[n] : -1;
}

// ---------------------------------------------------------------------------
// K2: embedding-bag. One block per output row; thread d owns column d so each
// gathered 1KB emb row is fully coalesced. Indices are block-uniform. Non-pad
// codes form a prefix of each row and emb[pad]==0, so break at the first pad:
// 85/60 gathers instead of 4000 (~180 MB of L2 traffic avoided).
// Blocks 0..22  -> v_emb[t]  from full_seq rows 0..22
// Blocks 23..45 -> cur[t]    from cur_seq  rows 1..23
// ---------------------------------------------------------------------------
__global__ __launch_bounds__(D_)
void embed_sum_kernel(const int* __restrict__ full_seq,
                      const int* __restrict__ cur_seq,
                      const float* __restrict__ emb,
                      float* __restrict__ v_emb, float* __restrict__ cur)
{
    const int d = threadIdx.x;
    const int b = blockIdx.x;
    const int* s;
    float* out;
    if (b < NT_) { s = full_seq + b * ILEN_;            out = v_emb + b * D_; }
    else         { s = cur_seq + (b - NT_ + 1) * ILEN_; out = cur + (b - NT_) * D_; }
    float acc = 0.0f;
    for (int j = 0; j < ILEN_; ++j) {
        int idx = __builtin_amdgcn_readfirstlane(s[j]);  // block-uniform
        if (idx == ILEN_) break;          // pads are a suffix; emb[pad] == 0
        acc += emb[idx * D_ + d];
    }
    out[d] = acc;
}

// ---------------------------------------------------------------------------
// K3/K5/K7: f32 WMMA GEMM   out[M,N] (+epilogue)  =  A[Mp,K] @ W[Np,K]^T + b
// One wave per 16x16 tile; K-loop of V_WMMA_F32_16X16X4_F32 (exact f32 math).
// PRECONDITION: A has >= 16-aligned padded rows (zeros), W has 16-aligned
// padded rows (zeros) -> the hot loop is two unconditional global_load_b64
// per step, zero EXEC-mask traffic. Fragment layouts per CDNA5 ISA 7.12.2:
//   A 16x4 : lanes 0-15 M=lane, VGPRs {K0,K0+1}; lanes 16-31 {K0+2,K0+3}
//   B 4x16 : lanes 0-15 N=lane, same K striping
//   C/D    : VGPR v holds M=v (lanes 0-15) / M=v+8 (lanes 16-31), N=lane&15
// mode 0: out = acc + bias
// mode 1: out = tanh(acc + bias) * extra     (attn_h * v_emb, fused)
// mode 2: out = sigmoid(acc + bias)          (pred)
// ---------------------------------------------------------------------------
__global__ __launch_bounds__(128)
void wmma_gemm_nt_kernel(const float* __restrict__ A, int lda, int M,
                         const float* __restrict__ W, int ldw, int N,
                         const float* __restrict__ bias,
                         float* __restrict__ out, int ldo,
                         const float* __restrict__ extra,
                         int K, int mode)
{
    const int lane   = threadIdx.x & 31;
    const int wave   = blockIdx.x * (blockDim.x >> 5) + (threadIdx.x >> 5);
    const int ntiles = (N + 15) >> 4;
    const int mtiles = (M + 15) >> 4;
    if (wave >= mtiles * ntiles) return;
    const int m0 = (wave / ntiles) << 4;
    const int n0 = (wave - (wave / ntiles) * ntiles) << 4;

    const int half = lane >> 4;          // selects K sub-pair
    const int l16  = lane & 15;
    // Padded operands: no row/col guards needed for loads.
    const float* ap = A + (m0 + l16) * lda + (half << 1);
    const float* wp = W + (n0 + l16) * ldw + (half << 1);

    v8f acc = {};
#pragma unroll 8
    for (int k0 = 0; k0 < K; k0 += 4) {
        const v2f a = *(const v2f*)(ap + k0);   // 8B-aligned, unconditional
        const v2f b = *(const v2f*)(wp + k0);
        acc = __builtin_amdgcn_wmma_f32_16x16x4_f32(
                  false, a, false, b, (short)0, acc, false, false);
    }

    // Cold epilogue: bounds guards only here, after all WMMA issues.
    const int col = n0 + l16;
    if (col >= N) return;
    const float bv = bias[col];
#pragma unroll
    for (int v = 0; v < 8; ++v) {
        const int row = m0 + v + (half << 3);
        if (row >= M) continue;
        float val = acc[v] + bv;
        if (mode == 1)      val = tanhf(val) * extra[row * ldo + col];
        else if (mode == 2) val = 1.0f / (1.0f + expf(-val));
        out[row * ldo + col] = val;
    }
}

// ---------------------------------------------------------------------------
// K4: GRU recurrence (the serial critical path). Single workgroup, 768 threads
// (24 wave32s on one WGP). h lives in LDS (broadcast reads), gh in LDS. Each
// step: 768 threads each do one dot-256 against an L2-resident w_hh row, then
// threads 0..255 apply the gate math. gi was precomputed by WMMA GEMM.
// PyTorch gate order (r, z, n).
// ---------------------------------------------------------------------------
__global__ __launch_bounds__(G3_)
void gru_recurrence_kernel(const float* __restrict__ gi,    // [NT, 768]
                           const float* __restrict__ w_hh,  // [768, 256]
                           const float* __restrict__ b_hh,  // [768]
                           float* __restrict__ hs)          // [MP, 256]
{
    __shared__ float h[D_];
    __shared__ float gh[G3_];
    const int tid = threadIdx.x;
    if (tid < D_) h[tid] = 0.0f;
    __syncthreads();

    for (int t = 0; t < NT_; ++t) {
        float dot = b_hh[tid];
        const float* wr = w_hh + tid * D_;
#pragma unroll 8
        for (int k = 0; k < D_; ++k) dot += h[k] * wr[k];
        gh[tid] = dot;
        __syncthreads();

        float hnew = 0.0f;
        if (tid < D_) {
            const float* git = gi + t * G3_;
            float r = 1.0f / (1.0f + expf(-(git[tid]          + gh[tid])));
            float z = 1.0f / (1.0f + expf(-(git[D_ + tid]     + gh[D_ + tid])));
            float n = tanhf(git[2 * D_ + tid] + r * gh[2 * D_ + tid]);
            hnew = (1.0f - z) * n + z * h[tid];
        }
        __syncthreads();
        if (tid < D_) { h[tid] = hnew; hs[t * D_ + tid] = hnew; }
        __syncthreads();
    }
}

// ---------------------------------------------------------------------------
// K6: c = cumsum_t(s) ; x = c + cur. attn_g == 1 (softmax over a size-1 axis),
// so GRU1 / w_a1 / b_a1 are dead code and never computed.
// ---------------------------------------------------------------------------
__global__ __launch_bounds__(D_)
void cumsum_add_kernel(const float* __restrict__ s,
                       const float* __restrict__ cur,
                       float* __restrict__ x)
{
    const int d = threadIdx.x;
    float acc = 0.0f;
    for (int t = 0; t < NT_; ++t) {
        acc += s[t * D_ + d];
        x[t * D_ + d] = acc + cur[t * D_ + d];
    }
}

// ---------------------------------------------------------------------------
// K8: loss = sum_t mean_n BCE = (1/V2) * sum over all 23*500 elements.
// Single-block deterministic tree reduction.
// ---------------------------------------------------------------------------
__global__ __launch_bounds__(512)
void bce_loss_kernel(const float* __restrict__ y1,
                     const float* __restrict__ pred,
                     float* __restrict__ loss)
{
    __shared__ float red[512];
    const int tid = threadIdx.x;
    float s = 0.0f;
    for (int i = tid; i < NT_ * V2_; i += 512) {
        float p = pred[i];
        s += (y1[i] > 0.5f) ? -logf(p) : -log1pf(-p);
    }
    red[tid] = s;
    __syncthreads();
    for (int off = 256; off > 0; off >>= 1) {
        if (tid < off) red[tid] += red[tid + off];
        __syncthreads();
    }
    if (tid == 0) loss[0] = red[0] * (1.0f / (float)V2_);
}

// ---------------------------------------------------------------------------
// Launch
// ---------------------------------------------------------------------------
extern "C" void kernel_launch(void* const* d_in, const int* in_sizes, int n_in,
                              void* d_out, int out_size, void* d_ws, size_t ws_size,
                              hipStream_t stream)
{
    (void)in_sizes; (void)n_in; (void)out_size; (void)ws_size;

    const int*   full_seq = (const int*)  d_in[0];
    const int*   cur_seq  = (const int*)  d_in[1];
    const int*   med      = (const int*)  d_in[2];
    const float* emb      = (const float*)d_in[3];
    // d_in[4..7]  = w_ih1, w_hh1, b_ih1, b_hh1  (dead: attn_g == 1)
    const float* w_ih2    = (const float*)d_in[8];
    const float* w_hh2    = (const float*)d_in[9];
    const float* b_ih2    = (const float*)d_in[10];
    const float* b_hh2    = (const float*)d_in[11];
    // d_in[12..13] = w_a1, b_a1 (dead)
    const float* w_a2     = (const float*)d_in[14];
    const float* b_a2     = (const float*)d_in[15];
    const float* w_out    = (const float*)d_in[16];
    const float* b_out    = (const float*)d_in[17];

    // Output layout: y1 | y2(int) | pred | loss
    float* y1   = (float*)d_out;
    int*   y2   = (int*)((float*)d_out + NT_ * V2_);
    float* pred = (float*)d_out + 2 * NT_ * V2_;
    float* loss = (float*)d_out + 3 * NT_ * V2_;

    // Workspace layout (floats; all offsets 1KB-aligned)
    float* ws       = (float*)d_ws;
    float* v_emb    = ws;                          // [32, 256] (rows 23.. zero)
    float* cur      = v_emb    + MP_ * D_;         // [23, 256]
    float* gi2      = cur      + MP_ * D_;         // [23, 768]
    float* h2       = gi2      + MP_ * G3_;        // [32, 256] (rows 23.. zero)
    float* s_buf    = h2       + MP_ * D_;         // [23, 256]
    float* x_buf    = s_buf    + MP_ * D_;         // [32, 256] (rows 23.. zero)
    float* wout_pad = x_buf    + MP_ * D_;         // [512, 256] (rows 500.. zero)

    // K0: padding init (A-row padding + padded w_out copy)
    pad_zero_kernel<<<27, D_, 0, stream>>>(v_emb, h2, x_buf);
    pad_wout_kernel<<<V2P_, D_, 0, stream>>>(w_out, wout_pad);

    // K1: labels (independent of everything else)
    build_labels_kernel<<<(NT_ * V2_ + 255) / 256, 256, 0, stream>>>(med, y1, y2);

    // K2: embedding bags -> v_emb, cur
    embed_sum_kernel<<<2 * NT_, D_, 0, stream>>>(full_seq, cur_seq, emb, v_emb, cur);

    // K3: gi2 = v_emb @ w_ih2^T + b_ih2   [23,768]; tiles = 2*48 = 96 waves
    wmma_gemm_nt_kernel<<<24, 128, 0, stream>>>(v_emb, D_, NT_, w_ih2, D_, G3_,
                                                b_ih2, gi2, G3_, nullptr, D_, 0);

    // K4: serial GRU2 recurrence -> h2
    gru_recurrence_kernel<<<1, G3_, 0, stream>>>(gi2, w_hh2, b_hh2, h2);

    // K5: s = tanh(h2 @ w_a2^T + b_a2) * v_emb   [23,256]; 32 waves
    wmma_gemm_nt_kernel<<<8, 128, 0, stream>>>(h2, D_, NT_, w_a2, D_, D_,
                                               b_a2, s_buf, D_, v_emb, D_, 1);

    // K6: x = cumsum(s) + cur
    cumsum_add_kernel<<<1, D_, 0, stream>>>(s_buf, cur, x_buf);

    // K7: pred = sigmoid(x @ wout_pad^T + b_out)  [23,500]; tiles = 2*32 = 64
    wmma_gemm_nt_kernel<<<16, 128, 0, stream>>>(x_buf, D_, NT_, wout_pad, D_, V2_,
                                                b_out, pred, V2_, nullptr, D_, 2);

    // K8: scalar BCE loss
    bce_loss_kernel<<<1, 512, 0, stream>>>(y1, pred, loss);
}